// GAT_14310831030549
// MI455X (gfx1250) — compile-verified
//
#include <hip/hip_runtime.h>
#include <hip/hip_bf16.h>

// ---------------------------------------------------------------------------
// GAT forward (2 layers) for MI455X / gfx1250, wave32.
//   Layer1: h = x@W1 (WMMA f32 16x16x4), per-edge softmax attention (8 heads x
//           8 ch), scatter aggregation, bias+ELU.
//   Layer2: 64->2 linear, 1-head attention, aggregation, bias, log_softmax.
// Edge phase is bandwidth/atomic bound; node arrays (<16 MB) are L2-resident
// (192 MB L2). All edge passes are vectorized with float4 so the dominant
// aggregation pass moves ~640 B of loads + 256 B of atomics per edge instead
// of ~1.8 KB. The compute-dense piece (x@W1) uses the CDNA5 matrix pipe via
// __builtin_amdgcn_wmma_f32_16x16x4_f32 (full f32 precision).
// ---------------------------------------------------------------------------

#define GAT_H   8
#define GAT_C   8
#define GAT_HC  64
#define GAT_FIN 48
#define GAT_NC  2
#define NEG_SLOPE 0.2f
#define FLOAT_NEG_MAX (-3.402823466e38f)

typedef __attribute__((ext_vector_type(2))) float v2f;
typedef __attribute__((ext_vector_type(8))) float v8f;

// Float atomic max via sign-split integer atomics (NaN-free inputs).
__device__ __forceinline__ void atomic_max_f32(float* addr, float val) {
  if (__float_as_int(val) >= 0) {
    atomicMax((int*)addr, __float_as_int(val));           // non-negative path
  } else {
    atomicMin((unsigned int*)addr, __float_as_uint(val)); // negative path
  }
}

__device__ __forceinline__ void edge_nodes(const int* __restrict__ ei,
                                           long long e, long long E,
                                           int& s, int& d) {
  if (e < E) { s = ei[e]; d = ei[E + e]; }
  else       { s = d = (int)(e - E); }   // synthesized self loop
}

__device__ __forceinline__ float leaky(float v) {
  return v >= 0.0f ? v : NEG_SLOPE * v;
}

// ---------------------------------------------------------------------------
// Kernel 1: h1 = x @ W1   (N x 48) @ (48 x 64) -> (N x 64), f32 WMMA.
// One wave per 16-row tile; 4 accumulators cover all 64 output columns.
// A-frag (16x4 f32): lane m=lane&15 is row M, lane-half selects K pair.
// B-frag (4x16 f32): lane m is column N, lane-half selects K pair likewise.
// C/D (16x16 f32): VGPR r holds row r + 8*half, column = lane&15.
// EXEC is all-ones (32-thread block, no divergence) as WMMA requires.
// ---------------------------------------------------------------------------
__global__ void __launch_bounds__(32)
k_gemm1_wmma(const float* __restrict__ X, const float* __restrict__ W,
             float* __restrict__ Hout) {
  const int row0 = blockIdx.x * 16;
  const int lane = threadIdx.x & 31;
  const int m    = lane & 15;
  const int half = lane >> 4;

  v8f acc0 = {}, acc1 = {}, acc2 = {}, acc3 = {};

  for (int kb = 0; kb < GAT_FIN; kb += 4) {
    const int ka = kb + half * 2;
    const float* xr = X + (size_t)(row0 + m) * GAT_FIN + ka;
    v2f a; a.x = xr[0]; a.y = xr[1];

    const float* w0 = W + (size_t)ka * GAT_HC;  // row ka of W1
    const float* w1 = w0 + GAT_HC;              // row ka+1
    v2f b0, b1, b2, b3;
    b0.x = w0[m +  0]; b0.y = w1[m +  0];
    b1.x = w0[m + 16]; b1.y = w1[m + 16];
    b2.x = w0[m + 32]; b2.y = w1[m + 32];
    b3.x = w0[m + 48]; b3.y = w1[m + 48];

    acc0 = __builtin_amdgcn_wmma_f32_16x16x4_f32(false, a, false, b0, (short)0, acc0, false, false);
    acc1 = __builtin_amdgcn_wmma_f32_16x16x4_f32(false, a, false, b1, (short)0, acc1, false, false);
    acc2 = __builtin_amdgcn_wmma_f32_16x16x4_f32(false, a, false, b2, (short)0, acc2, false, false);
    acc3 = __builtin_amdgcn_wmma_f32_16x16x4_f32(false, a, false, b3, (short)0, acc3, false, false);
  }

  float* o = Hout + (size_t)row0 * GAT_HC;
#pragma unroll
  for (int r = 0; r < 8; ++r) {
    const int row = r + half * 8;
    float* orow = o + (size_t)row * GAT_HC;
    orow[m +  0] = acc0[r];
    orow[m + 16] = acc1[r];
    orow[m + 32] = acc2[r];
    orow[m + 48] = acc3[r];
  }
}

// Kernel 2: per-node attention logits e_src/e_dst, layer 1.  idx = node*8+head
__global__ void k_att1(const float* __restrict__ Hf, const float* __restrict__ as,
                       const float* __restrict__ ad, float* __restrict__ es,
                       float* __restrict__ ed, int n) {
  int idx = blockIdx.x * blockDim.x + threadIdx.x;
  if (idx >= n * GAT_H) return;
  int node = idx >> 3, hd = idx & 7;
  const float4* hp = (const float4*)(Hf + (size_t)node * GAT_HC + hd * GAT_C);
  const float4* ap = (const float4*)(as + hd * GAT_C);
  const float4* bp = (const float4*)(ad + hd * GAT_C);
  float4 h0 = hp[0], h1 = hp[1];
  float4 a0 = ap[0], a1 = ap[1];
  float4 b0 = bp[0], b1 = bp[1];
  es[idx] = h0.x * a0.x + h0.y * a0.y + h0.z * a0.z + h0.w * a0.w +
            h1.x * a1.x + h1.y * a1.y + h1.z * a1.z + h1.w * a1.w;
  ed[idx] = h0.x * b0.x + h0.y * b0.y + h0.z * b0.z + h0.w * b0.w +
            h1.x * b1.x + h1.y * b1.y + h1.z * b1.z + h1.w * b1.w;
}

// Kernel 3: init layer-1 reducers (m=-FLT_MAX, den=0, agg=0). n64 = N*64
__global__ void k_init1(float* __restrict__ m, float* __restrict__ den,
                        float* __restrict__ agg, long long n8, long long n64) {
  long long idx = (long long)blockIdx.x * blockDim.x + threadIdx.x;
  if (idx < n64) agg[idx] = 0.f;
  if (idx < n8) { m[idx] = FLOAT_NEG_MAX; den[idx] = 0.f; }
}

// Kernel 4: segment max over incoming edges. One thread per edge, 8 heads via
// two float4 loads of each logit vector (64 B instead of 128 B per edge).
__global__ void k_edge1_max(const int* __restrict__ ei, long long E, long long ET,
                            const float* __restrict__ es, const float* __restrict__ ed,
                            float* __restrict__ m) {
  long long e = (long long)blockIdx.x * blockDim.x + threadIdx.x;
  if (e >= ET) return;
  int s, d; edge_nodes(ei, e, E, s, d);
  const float4* sp = (const float4*)(es + (size_t)s * GAT_H);
  const float4* dp = (const float4*)(ed + (size_t)d * GAT_H);
  float4 s0 = sp[0], s1 = sp[1], d0 = dp[0], d1 = dp[1];
  float v[8] = { s0.x + d0.x, s0.y + d0.y, s0.z + d0.z, s0.w + d0.w,
                 s1.x + d1.x, s1.y + d1.y, s1.z + d1.z, s1.w + d1.w };
  float* mp = m + (size_t)d * GAT_H;
#pragma unroll
  for (int h = 0; h < GAT_H; ++h) atomic_max_f32(&mp[h], leaky(v[h]));
}

// Kernel 5: segment sum of exp(e - m). One thread per edge, 8 heads.
__global__ void k_edge1_sum(const int* __restrict__ ei, long long E, long long ET,
                            const float* __restrict__ es, const float* __restrict__ ed,
                            const float* __restrict__ m, float* __restrict__ den) {
  long long e = (long long)blockIdx.x * blockDim.x + threadIdx.x;
  if (e >= ET) return;
  int s, d; edge_nodes(ei, e, E, s, d);
  const float4* sp = (const float4*)(es + (size_t)s * GAT_H);
  const float4* dp = (const float4*)(ed + (size_t)d * GAT_H);
  const float4* mp4 = (const float4*)(m + (size_t)d * GAT_H);
  float4 s0 = sp[0], s1 = sp[1], d0 = dp[0], d1 = dp[1];
  float4 m0 = mp4[0], m1 = mp4[1];
  float v[8] = { s0.x + d0.x, s0.y + d0.y, s0.z + d0.z, s0.w + d0.w,
                 s1.x + d1.x, s1.y + d1.y, s1.z + d1.z, s1.w + d1.w };
  float mv[8] = { m0.x, m0.y, m0.z, m0.w, m1.x, m1.y, m1.z, m1.w };
  float* dnp = den + (size_t)d * GAT_H;
#pragma unroll
  for (int h = 0; h < GAT_H; ++h)
    atomicAdd(&dnp[h], __expf(leaky(v[h]) - mv[h]));
}

// Kernel 6: aggregation  agg[dst,k] += alpha * h1[src,k].
// One thread per (edge, 4-channel group): idx = edge*16 + g, channels g*4..g*4+3
// all within head g>>1. float4 gather of features; 4 consecutive f32 atomics.
__global__ void k_edge1_agg(const int* __restrict__ ei, long long E, long long ET,
                            const float* __restrict__ es, const float* __restrict__ ed,
                            const float* __restrict__ m, const float* __restrict__ den,
                            const float* __restrict__ h1, float* __restrict__ agg) {
  long long idx = (long long)blockIdx.x * blockDim.x + threadIdx.x;
  if (idx >= ET * 16LL) return;
  long long e = idx >> 4;
  int g  = (int)(idx & 15);
  int k0 = g * 4;
  int hd = g >> 1;
  int s, d; edge_nodes(ei, e, E, s, d);
  float v = leaky(es[(size_t)s * GAT_H + hd] + ed[(size_t)d * GAT_H + hd]);
  float alpha = __expf(v - m[(size_t)d * GAT_H + hd]) / den[(size_t)d * GAT_H + hd];
  float4 hv = *(const float4*)(h1 + (size_t)s * GAT_HC + k0);
  float* ap = agg + (size_t)d * GAT_HC + k0;
  atomicAdd(&ap[0], alpha * hv.x);
  atomicAdd(&ap[1], alpha * hv.y);
  atomicAdd(&ap[2], alpha * hv.z);
  atomicAdd(&ap[3], alpha * hv.w);
}

// Kernel 7: in-place bias + ELU on agg (float4) -> becomes h2. n16 = N*16
__global__ void k_bias_elu(float4* __restrict__ agg, const float4* __restrict__ b,
                           long long n16) {
  long long idx = (long long)blockIdx.x * blockDim.x + threadIdx.x;
  if (idx >= n16) return;
  float4 v = agg[idx];
  float4 bb = b[idx & 15];
  v.x += bb.x; v.y += bb.y; v.z += bb.z; v.w += bb.w;
  v.x = v.x > 0.f ? v.x : (__expf(v.x) - 1.f);
  v.y = v.y > 0.f ? v.y : (__expf(v.y) - 1.f);
  v.z = v.z > 0.f ? v.z : (__expf(v.z) - 1.f);
  v.w = v.w > 0.f ? v.w : (__expf(v.w) - 1.f);
  agg[idx] = v;
}

// Kernel 8: layer-2 linear (64->2) + attention logits, one thread per node
__global__ void k_lin2(const float* __restrict__ h2, const float* __restrict__ W2,
                       const float* __restrict__ as2, const float* __restrict__ ad2,
                       float* __restrict__ hw, float* __restrict__ es2,
                       float* __restrict__ ed2, int n) {
  int node = blockIdx.x * blockDim.x + threadIdx.x;
  if (node >= n) return;
  const float4* hp = (const float4*)(h2 + (size_t)node * GAT_HC);
  float v0 = 0.f, v1 = 0.f;
#pragma unroll
  for (int q = 0; q < GAT_HC / 4; ++q) {
    float4 hv = hp[q];
    const float* wq = W2 + q * 4 * GAT_NC;
    v0 += hv.x * wq[0] + hv.y * wq[2] + hv.z * wq[4] + hv.w * wq[6];
    v1 += hv.x * wq[1] + hv.y * wq[3] + hv.z * wq[5] + hv.w * wq[7];
  }
  hw[(size_t)node * GAT_NC + 0] = v0;
  hw[(size_t)node * GAT_NC + 1] = v1;
  es2[node] = v0 * as2[0] + v1 * as2[1];
  ed2[node] = v0 * ad2[0] + v1 * ad2[1];
}

// Kernel 9: init layer-2 reducers
__global__ void k_init2(float* __restrict__ m, float* __restrict__ den,
                        float* __restrict__ agg, int n) {
  int idx = blockIdx.x * blockDim.x + threadIdx.x;
  if (idx < 2 * n) agg[idx] = 0.f;
  if (idx < n) { m[idx] = FLOAT_NEG_MAX; den[idx] = 0.f; }
}

// Kernel 10: layer-2 segment max (H=1)
__global__ void k_edge2_max(const int* __restrict__ ei, long long E, long long ET,
                            const float* __restrict__ es, const float* __restrict__ ed,
                            float* __restrict__ m) {
  long long e = (long long)blockIdx.x * blockDim.x + threadIdx.x;
  if (e >= ET) return;
  int s, d; edge_nodes(ei, e, E, s, d);
  float v = leaky(es[s] + ed[d]);
  atomic_max_f32(&m[d], v);
}

// Kernel 11: layer-2 exp-sum
__global__ void k_edge2_sum(const int* __restrict__ ei, long long E, long long ET,
                            const float* __restrict__ es, const float* __restrict__ ed,
                            const float* __restrict__ m, float* __restrict__ den) {
  long long e = (long long)blockIdx.x * blockDim.x + threadIdx.x;
  if (e >= ET) return;
  int s, d; edge_nodes(ei, e, E, s, d);
  float v = leaky(es[s] + ed[d]);
  atomicAdd(&den[d], __expf(v - m[d]));
}

// Kernel 12: layer-2 aggregation (2 channels per edge)
__global__ void k_edge2_agg(const int* __restrict__ ei, long long E, long long ET,
                            const float* __restrict__ es, const float* __restrict__ ed,
                            const float* __restrict__ m, const float* __restrict__ den,
                            const float* __restrict__ hw, float* __restrict__ agg) {
  long long e = (long long)blockIdx.x * blockDim.x + threadIdx.x;
  if (e >= ET) return;
  int s, d; edge_nodes(ei, e, E, s, d);
  float v = leaky(es[s] + ed[d]);
  float alpha = __expf(v - m[d]) / den[d];
  atomicAdd(&agg[(size_t)d * GAT_NC + 0], alpha * hw[(size_t)s * GAT_NC + 0]);
  atomicAdd(&agg[(size_t)d * GAT_NC + 1], alpha * hw[(size_t)s * GAT_NC + 1]);
}

// Kernel 13: bias + 2-class log_softmax -> d_out
__global__ void k_final(const float* __restrict__ agg, const float* __restrict__ b2,
                        float* __restrict__ out, int n) {
  int node = blockIdx.x * blockDim.x + threadIdx.x;
  if (node >= n) return;
  float v0 = agg[(size_t)node * GAT_NC + 0] + b2[0];
  float v1 = agg[(size_t)node * GAT_NC + 1] + b2[1];
  float mx = fmaxf(v0, v1);
  float lse = mx + __logf(__expf(v0 - mx) + __expf(v1 - mx));
  out[(size_t)node * GAT_NC + 0] = v0 - lse;
  out[(size_t)node * GAT_NC + 1] = v1 - lse;
}

static inline int grid_for(long long n, int b) { return (int)((n + b - 1) / b); }

extern "C" void kernel_launch(void* const* d_in, const int* in_sizes, int n_in,
                              void* d_out, int out_size, void* d_ws, size_t ws_size,
                              hipStream_t stream) {
  const float* x   = (const float*)d_in[0];
  const int*   ei  = (const int*)d_in[1];
  const float* W1  = (const float*)d_in[2];
  const float* as1 = (const float*)d_in[3];
  const float* ad1 = (const float*)d_in[4];
  const float* b1  = (const float*)d_in[5];
  const float* W2  = (const float*)d_in[6];
  const float* as2 = (const float*)d_in[7];
  const float* ad2 = (const float*)d_in[8];
  const float* b2  = (const float*)d_in[9];
  float* out = (float*)d_out;

  const int       N  = in_sizes[0] / GAT_FIN;            // 50000
  const long long E  = (long long)(in_sizes[1] / 2);     // 1.6M
  const long long ET = E + N;                            // + self loops

  // Workspace carve-up (~35 MB of f32); every sub-array 16 B aligned.
  float* p    = (float*)d_ws;
  float* h1   = p; p += (size_t)N * GAT_HC;   // layer-1 features
  float* agg1 = p; p += (size_t)N * GAT_HC;   // aggregation -> h2 after ELU
  float* es1  = p; p += (size_t)N * GAT_H;
  float* ed1  = p; p += (size_t)N * GAT_H;
  float* m1   = p; p += (size_t)N * GAT_H;
  float* dn1  = p; p += (size_t)N * GAT_H;
  float* hw2  = p; p += (size_t)N * GAT_NC;
  float* es2  = p; p += (size_t)N;
  float* ed2  = p; p += (size_t)N;
  float* m2   = p; p += (size_t)N;
  float* dn2  = p; p += (size_t)N;
  float* agg2 = p; p += (size_t)N * GAT_NC;

  const int B = 256;

  // ---- layer 1 ----
  k_gemm1_wmma<<<N / 16, 32, 0, stream>>>(x, W1, h1);
  k_att1<<<grid_for((long long)N * GAT_H, B), B, 0, stream>>>(h1, as1, ad1, es1, ed1, N);
  k_init1<<<grid_for((long long)N * GAT_HC, B), B, 0, stream>>>(m1, dn1, agg1,
                                                               (long long)N * GAT_H,
                                                               (long long)N * GAT_HC);
  k_edge1_max<<<grid_for(ET, B), B, 0, stream>>>(ei, E, ET, es1, ed1, m1);
  k_edge1_sum<<<grid_for(ET, B), B, 0, stream>>>(ei, E, ET, es1, ed1, m1, dn1);
  k_edge1_agg<<<grid_for(ET * 16LL, B), B, 0, stream>>>(ei, E, ET, es1, ed1, m1, dn1, h1, agg1);
  k_bias_elu<<<grid_for((long long)N * 16, B), B, 0, stream>>>((float4*)agg1, (const float4*)b1,
                                                               (long long)N * 16);

  // ---- layer 2 ----
  k_lin2<<<grid_for(N, B), B, 0, stream>>>(agg1, W2, as2, ad2, hw2, es2, ed2, N);
  k_init2<<<grid_for(2LL * N, B), B, 0, stream>>>(m2, dn2, agg2, N);
  k_edge2_max<<<grid_for(ET, B), B, 0, stream>>>(ei, E, ET, es2, ed2, m2);
  k_edge2_sum<<<grid_for(ET, B), B, 0, stream>>>(ei, E, ET, es2, ed2, m2, dn2);
  k_edge2_agg<<<grid_for(ET, B), B, 0, stream>>>(ei, E, ET, es2, ed2, m2, dn2, hw2, agg2);
  k_final<<<grid_for(N, B), B, 0, stream>>>(agg2, b2, out, N);
}